// MmdLoss_27075473834451
// MI455X (gfx1250) — compile-verified
//
#include <hip/hip_runtime.h>

typedef __attribute__((ext_vector_type(2))) float v2f;
typedef __attribute__((ext_vector_type(8))) float v8f;

// ---- workspace layout (float offsets) ----
#define WS_SUMX 0          // [8] per-batch sum of input
#define WS_SUMT 8          // [8] per-batch sum of target
#define WS_SUMQ 16         // [8] per-batch sum of q_raw
#define WS_SUMP 24         // [8] per-batch sum of p_raw
#define WS_LOSS 32         // [8] accumulated d' K d
#define WS_TH   40         // [0]=th_x [1]=th_t
#define WS_INVQ 48         // [8]
#define WS_INVP 56         // [8]
#define WS_KG   64                 // 128x128 Gaussian, 16384 floats
#define WS_QRAW (WS_KG + 16384)    // [8][128][128]
#define WS_PRAW (WS_QRAW + 131072) // [8][128][128]
#define WS_M1   (WS_PRAW + 131072) // [8][128][128]

#define BATCHES 8
#define L 262144           // 512*512
#define M 128              // pooled grid side
#define N 16384            // M*M

// ---------------- tiny setup kernels ----------------
__global__ void init_k(float* ws) {
    if (threadIdx.x < 64) ws[threadIdx.x] = 0.0f;
}

__global__ void kgen_k(float* ws) {
    int t = blockIdx.x * blockDim.x + threadIdx.x;
    if (t >= N) return;
    int a = t >> 7, b = t & 127;
    float d = (float)(a - b);
    ws[WS_KG + t] = expf(-(d * d) * (1.0f / 128.0f));   // exp(-D / (2*64))
}

// ---------------- pass 1: global sums for thresholds + area loss ----------------
__global__ void sum_k(const float* __restrict__ x, const float* __restrict__ t, float* ws) {
    __shared__ float sx[256], st[256];
    int g   = blockIdx.x * 256 + threadIdx.x;  // [0, 8*65536)
    int b   = g >> 16;                          // batch (constant per block)
    int idx = (g & 65535) << 2;                 // float4 index within batch
    const float4 xv = *(const float4*)(x + b * L + idx);
    const float4 tv = *(const float4*)(t + b * L + idx);
    sx[threadIdx.x] = xv.x + xv.y + xv.z + xv.w;
    st[threadIdx.x] = tv.x + tv.y + tv.z + tv.w;
    __syncthreads();
    for (int s = 128; s > 0; s >>= 1) {
        if (threadIdx.x < s) {
            sx[threadIdx.x] += sx[threadIdx.x + s];
            st[threadIdx.x] += st[threadIdx.x + s];
        }
        __syncthreads();
    }
    if (threadIdx.x == 0) {
        atomicAdd(&ws[WS_SUMX + b], sx[0]);
        atomicAdd(&ws[WS_SUMT + b], st[0]);
    }
}

__global__ void prepth_k(float* ws) {
    if (threadIdx.x == 0) {
        float sx = 0.0f, st = 0.0f;
        for (int b = 0; b < BATCHES; ++b) { sx += ws[WS_SUMX + b]; st += ws[WS_SUMT + b]; }
        // th = max(mean_all * h*w / C, 0.01) = max(total / (B*C), 0.01)
        ws[WS_TH + 0] = fmaxf(sx * (1.0f / (8.0f * 500.0f)), 0.01f);
        ws[WS_TH + 1] = fmaxf(st * (1.0f / (8.0f * 100.0f)), 0.01f);
    }
}

// ---------------- pass 2: fused threshold-mask + 4x4 max/avg pooling ----------------
__global__ void pool_k(const float* __restrict__ x,  const float* __restrict__ t,
                       const float* __restrict__ ux, const float* __restrict__ ut,
                       float* ws) {
    __shared__ float sq[256], sp[256];
    int g = blockIdx.x * 256 + threadIdx.x;    // [0, 8*16384)
    int b = g >> 14;                            // constant per block (16384 % 256 == 0)
    int n = g & (N - 1);
    int i = n >> 7, j = n & 127;
    float thx = ws[WS_TH + 0], tht = ws[WS_TH + 1];
    int base = b * L + (i * 4) * 512 + j * 4;

    float sxa = 0.0f, sta = 0.0f;
    int selx = 0, selt = 0;
#pragma unroll
    for (int r = 0; r < 4; ++r) {
        const float4 xv  = *(const float4*)(x  + base + r * 512);
        const float4 tv  = *(const float4*)(t  + base + r * 512);
        const float4 uxv = *(const float4*)(ux + base + r * 512);
        const float4 utv = *(const float4*)(ut + base + r * 512);
        sxa += xv.x + xv.y + xv.z + xv.w;
        sta += tv.x + tv.y + tv.z + tv.w;
        selx |= (int)(xv.x > uxv.x * thx) | (int)(xv.y > uxv.y * thx) |
                (int)(xv.z > uxv.z * thx) | (int)(xv.w > uxv.w * thx);
        selt |= (int)(tv.x > utv.x * tht) | (int)(tv.y > utv.y * tht) |
                (int)(tv.z > utv.z * tht) | (int)(tv.w > utv.w * tht);
    }
    float qv = selx ? sxa * (1.0f / 16.0f) : 0.0f;  // avg-pool gated by max-pooled mask
    float pv = selt ? sta * (1.0f / 16.0f) : 0.0f;
    ws[WS_QRAW + b * N + n] = qv;
    ws[WS_PRAW + b * N + n] = pv;

    sq[threadIdx.x] = qv;
    sp[threadIdx.x] = pv;
    __syncthreads();
    for (int s = 128; s > 0; s >>= 1) {
        if (threadIdx.x < s) {
            sq[threadIdx.x] += sq[threadIdx.x + s];
            sp[threadIdx.x] += sp[threadIdx.x + s];
        }
        __syncthreads();
    }
    if (threadIdx.x == 0) {
        atomicAdd(&ws[WS_SUMQ + b], sq[0]);
        atomicAdd(&ws[WS_SUMP + b], sp[0]);
    }
}

__global__ void prepn_k(float* ws) {
    int b = threadIdx.x;
    if (b < BATCHES) {
        ws[WS_INVQ + b] = 1.0f / ws[WS_SUMQ + b];
        ws[WS_INVP + b] = 1.0f / ws[WS_SUMP + b];
    }
}

// ---------------- WMMA GEMM 1: M1 = Kg * Dm  (per batch, 128x128x128 f32) ----------------
// One wave (32 lanes) per 16x16 output tile; V_WMMA_F32_16X16X4_F32 over K.
__global__ void __launch_bounds__(32) gemm1_k(float* ws) {
    int bid  = blockIdx.x;           // [0, 8*64)
    int b    = bid >> 6;
    int tile = bid & 63;
    int i0 = (tile >> 3) * 16;
    int j0 = (tile & 7)  * 16;
    int lane = threadIdx.x;
    int m  = lane & 15;              // row within tile (A) / col within tile (B,C,D)
    int hi = lane >> 4;              // lane half selector

    const float* Kg = ws + WS_KG;
    const float* Q  = ws + WS_QRAW + b * N;
    const float* P  = ws + WS_PRAW + b * N;
    float invq = ws[WS_INVQ + b], invp = ws[WS_INVP + b];

    const float* Arow = Kg + (i0 + m) * M;   // A row for this lane
    int col = j0 + m;
    v8f c = {};
    for (int k0 = 0; k0 < M; k0 += 4) {
        int k = k0 + hi * 2;
        v2f a;                                 // A 16x4: VGPR0={K0,K2}, VGPR1={K1,K3}
        a.x = Arow[k];
        a.y = Arow[k + 1];
        int r0 = k * M + col;                  // B 4x16: rows K striped across lanes
        v2f bb;
        bb.x = Q[r0] * invq - P[r0] * invp;          // Dm[k  ][col]
        bb.y = Q[r0 + M] * invq - P[r0 + M] * invp;  // Dm[k+1][col]
        c = __builtin_amdgcn_wmma_f32_16x16x4_f32(
                false, a, false, bb, (short)0, c, false, false);
    }
    float* M1 = ws + WS_M1 + b * N;
#pragma unroll
    for (int v = 0; v < 8; ++v) {
        int row = i0 + v + hi * 8;             // C/D layout: VGPR v -> rows v / v+8
        M1[row * M + col] = c[v];
    }
}

// ---------------- WMMA GEMM 2 + dot: loss[b] += <Dm, M1 * Kg> ----------------
__global__ void __launch_bounds__(32) gemm2_k(float* ws) {
    int bid  = blockIdx.x;
    int b    = bid >> 6;
    int tile = bid & 63;
    int i0 = (tile >> 3) * 16;
    int j0 = (tile & 7)  * 16;
    int lane = threadIdx.x;
    int m  = lane & 15;
    int hi = lane >> 4;

    const float* Kg = ws + WS_KG;
    const float* M1 = ws + WS_M1 + b * N;
    const float* Q  = ws + WS_QRAW + b * N;
    const float* P  = ws + WS_PRAW + b * N;
    float invq = ws[WS_INVQ + b], invp = ws[WS_INVP + b];

    const float* Arow = M1 + (i0 + m) * M;
    int col = j0 + m;
    v8f c = {};
    for (int k0 = 0; k0 < M; k0 += 4) {
        int k = k0 + hi * 2;
        v2f a;
        a.x = Arow[k];
        a.y = Arow[k + 1];
        v2f bb;
        bb.x = Kg[k * M + col];
        bb.y = Kg[(k + 1) * M + col];
        c = __builtin_amdgcn_wmma_f32_16x16x4_f32(
                false, a, false, bb, (short)0, c, false, false);
    }
    // dot C-fragment against Dm tile
    float s = 0.0f;
#pragma unroll
    for (int v = 0; v < 8; ++v) {
        int idx = (i0 + v + hi * 8) * M + col;
        float dm = Q[idx] * invq - P[idx] * invp;
        s += dm * c[v];
    }
    for (int off = 16; off > 0; off >>= 1) s += __shfl_down(s, off);
    if (lane == 0) atomicAdd(&ws[WS_LOSS + b], s);
}

// ---------------- finalize ----------------
__global__ void final_k(const float* __restrict__ ws, float* __restrict__ out) {
    int b = threadIdx.x;
    if (b < BATCHES) {
        float dA   = (ws[WS_SUMX + b] - ws[WS_SUMT + b]) * (1.0f / 16.0f);
        float area = dA * dA * (1.0f / (float)L);
        out[b] = 0.5f * ws[WS_LOSS + b] + area;
    }
}

extern "C" void kernel_launch(void* const* d_in, const int* in_sizes, int n_in,
                              void* d_out, int out_size, void* d_ws, size_t ws_size,
                              hipStream_t stream) {
    const float* x  = (const float*)d_in[0];   // input
    const float* t  = (const float*)d_in[1];   // target
    const float* ux = (const float*)d_in[2];   // u_input
    const float* ut = (const float*)d_in[3];   // u_target
    float* out = (float*)d_out;                // [8]
    float* ws  = (float*)d_ws;                 // needs ~1.57 MB

    init_k  <<<1,    64,  0, stream>>>(ws);
    kgen_k  <<<64,   256, 0, stream>>>(ws);
    sum_k   <<<2048, 256, 0, stream>>>(x, t, ws);
    prepth_k<<<1,    32,  0, stream>>>(ws);
    pool_k  <<<512,  256, 0, stream>>>(x, t, ux, ut, ws);
    prepn_k <<<1,    32,  0, stream>>>(ws);
    gemm1_k <<<512,  32,  0, stream>>>(ws);
    gemm2_k <<<512,  32,  0, stream>>>(ws);
    final_k <<<1,    32,  0, stream>>>(ws, out);
}